// ProbaRanking_7834020348598
// MI455X (gfx1250) — compile-verified
//
#include <hip/hip_runtime.h>
#include <stdint.h>

typedef __attribute__((ext_vector_type(4))) unsigned int u32x4;
typedef __attribute__((ext_vector_type(4))) int          i32x4;
typedef __attribute__((ext_vector_type(8))) int          i32x8;
typedef __attribute__((ext_vector_type(2))) float        v2f;
typedef __attribute__((ext_vector_type(8))) float        v8f;

#ifndef __has_builtin
#define __has_builtin(x) 0
#endif

#define KN 8192            // row length
#define KT 1024            // threads per block (32 wave32 waves)
#define KE (KN / KT)       // 8 bins / elements per thread

__device__ __forceinline__ void wait_tensorcnt0() {
#if __has_builtin(__builtin_amdgcn_s_wait_tensorcnt)
  __builtin_amdgcn_s_wait_tensorcnt(0);
#else
  asm volatile("s_wait_tensorcnt 0" ::: "memory");
#endif
}

// Build a 1-row (8192 x f32) tile descriptor: D# group0 + group1.
__device__ __forceinline__ void tdm_row_desc(const void* gptr, const void* lptr,
                                             u32x4& g0, i32x8& g1) {
  unsigned long long ga = (unsigned long long)(uintptr_t)gptr;
  unsigned ldsa = (unsigned)(uintptr_t)lptr;            // low 32 bits = LDS offset
  g0 = { 1u,                                            // count=1, user descriptor
         ldsa,                                          // lds_addr
         (unsigned)ga,                                  // global_addr[31:0]
         ((unsigned)((ga >> 32) & 0x01FFFFFFu)) | 0x80000000u }; // addr[56:32]|type=2
  g1 = { (int)(2u << 16),                               // data_size = 4 bytes
         (int)(0x2000u << 16),                          // tensor_dim0 = 8192 (lo16)
         (int)(1u << 16),                               // dim0 hi=0, tensor_dim1 = 1
         (int)(0x2000u << 16),                          // tile_dim0 = 8192
         1,                                             // tile_dim1 = 1
         0x2000, 0, 0 };                                // tensor_dim0_stride = 8192
}

__global__ __launch_bounds__(KT) void ProbaRanking_7834020348598_kernel(
    const float* __restrict__ proba, const int* __restrict__ total,
    float* __restrict__ out) {
  __shared__ float              s_vals[KN];   // 32 KB : row values (TDM load target)
  __shared__ unsigned           s_hist[KN];   // 32 KB : hist -> offsets -> out stage
  __shared__ unsigned long long s_keys[KN];   // 64 KB : (~fbits,idx) rank-ordered
  float* s_partf  = reinterpret_cast<float*>(s_keys);   // 1024 scan partials (alias)
  float* s_rowtot = s_partf + KT;                       // 64 row totals (alias)

  const int    row   = blockIdx.x;
  const int    tid   = threadIdx.x;
  const size_t rbase = (size_t)row * KN;

  // ---- Stage 1: DMA the row into LDS with the Tensor Data Mover ----------
#if __has_builtin(__builtin_amdgcn_tensor_load_to_lds)
  if (tid < 32) {
    u32x4 g0; i32x8 g1;
    tdm_row_desc(proba + rbase, &s_vals[0], g0, g1);
    i32x4 z4 = {0, 0, 0, 0};
#if __clang_major__ >= 23
    i32x8 z8 = {0, 0, 0, 0, 0, 0, 0, 0};
    __builtin_amdgcn_tensor_load_to_lds(g0, g1, z4, z4, z8, 0);
#else
    __builtin_amdgcn_tensor_load_to_lds(g0, g1, z4, z4, 0);
#endif
  }
#endif

  // Zero the histogram while the DMA is in flight.
  for (int j = tid; j < KN; j += KT) s_hist[j] = 0u;

#if __has_builtin(__builtin_amdgcn_tensor_load_to_lds)
  if (tid < 32) wait_tensorcnt0();
#else
  for (int j = tid; j < KN; j += KT) s_vals[j] = proba[rbase + j];
#endif
  __syncthreads();

  // ---- Stage 2: bucket histogram (bucket bb ascending == value descending)
  for (int j = tid; j < KN; j += KT) {
    float f = s_vals[j];
    int b = (int)(f * 8192.0f);
    b = b < 0 ? 0 : (b > 8191 ? 8191 : b);
    atomicAdd(&s_hist[8191 - b], 1u);
  }
  __syncthreads();

  // ---- Stage 3: exclusive prefix sum over 8192 bins ----------------------
  // Per-thread serial reduce of 8 bins, then scan of the 1024 partials.
  unsigned loc[KE];
  unsigned mysum = 0;
  {
    const int base = tid * KE;
#pragma unroll
    for (int i = 0; i < KE; ++i) { loc[i] = s_hist[base + i]; mysum += loc[i]; }
  }
  s_partf[tid] = (float)mysum;   // counts <= 8192: exact in f32
  __syncthreads();

#if __has_builtin(__builtin_amdgcn_wmma_f32_16x16x4_f32)
  // Matrix-engine scan: each of 4 waves scans one 256-partial tile as a
  // 16x16 row-major X; inclusive row prefix P = X * U  (U = upper-tri ones)
  // via a K-chained V_WMMA_F32_16X16X4_F32.
  if (tid < 128) {
    const int wave = tid >> 5;
    const int lane = tid & 31;
    const int hi   = lane >> 4;           // half-wave select (K pairs / M+8)
    const int m    = lane & 15;           // A row / B,D column
    float* X = &s_partf[wave * 256];
    v8f acc = {0.f, 0.f, 0.f, 0.f, 0.f, 0.f, 0.f, 0.f};
#pragma unroll
    for (int g = 0; g < 4; ++g) {
      const int kb = 4 * g + 2 * hi;
      v2f a = { X[m * 16 + kb], X[m * 16 + kb + 1] };   // A = X slice (16x4)
      v2f b = { (kb     <= m) ? 1.0f : 0.0f,            // B = U slice (4x16)
                (kb + 1 <= m) ? 1.0f : 0.0f };
      acc = __builtin_amdgcn_wmma_f32_16x16x4_f32(false, a, false, b,
                                                  (short)0, acc, false, false);
    }
#pragma unroll
    for (int i = 0; i < 8; ++i) {         // D layout: VGPR i -> row i (+8 hi half)
      const int r = i + 8 * hi;
      X[r * 16 + m] = acc[i];             // inclusive prefix within 16-col row
      if (m == 15) s_rowtot[wave * 16 + r] = acc[i];    // row total
    }
  }
  __syncthreads();
  if (tid == 0) {                          // exclusive scan of the 64 row totals
    float run = 0.f;
    for (int i = 0; i < 64; ++i) { float v = s_rowtot[i]; s_rowtot[i] = run; run += v; }
  }
  __syncthreads();
  {
    const int col = tid & 15;
    float basef = s_rowtot[tid >> 4] + (col ? s_partf[tid - 1] : 0.0f);
    unsigned run = (unsigned)basef;
    const int base = tid * KE;
#pragma unroll
    for (int i = 0; i < KE; ++i) { s_hist[base + i] = run; run += loc[i]; }
  }
  __syncthreads();
#else
  // Fallback: Hillis-Steele over the 1024 partials.
  for (int ofs = 1; ofs < KT; ofs <<= 1) {
    float v = (tid >= ofs) ? s_partf[tid - ofs] : 0.0f;
    __syncthreads();
    s_partf[tid] += v;
    __syncthreads();
  }
  {
    unsigned run = (tid == 0) ? 0u : (unsigned)s_partf[tid - 1];
    const int base = tid * KE;
#pragma unroll
    for (int i = 0; i < KE; ++i) { s_hist[base + i] = run; run += loc[i]; }
  }
  __syncthreads();
#endif

  // ---- Stage 4: scatter (key = ~fbits:idx) into rank-ordered segments ----
  for (int j = tid; j < KN; j += KT) {
    float f = s_vals[j];
    int b = (int)(f * 8192.0f);
    b = b < 0 ? 0 : (b > 8191 ? 8191 : b);
    unsigned bb  = (unsigned)(8191 - b);
    unsigned pos = atomicAdd(&s_hist[bb], 1u);
    unsigned fb  = __float_as_uint(f);
    s_keys[pos]  = ((unsigned long long)(~fb) << 32) | (unsigned)j;
  }
  __syncthreads();

  // ---- Stage 5: per-bucket insertion micro-sort (expected ~1 elem/bucket)
  for (int bb = tid; bb < KN; bb += KT) {
    unsigned beg = (bb == 0) ? 0u : s_hist[bb - 1];  // post-scatter: hist = seg end
    unsigned end = s_hist[bb];
    for (unsigned u = beg + 1; u < end; ++u) {
      unsigned long long kv = s_keys[u];
      unsigned v = u;
      while (v > beg && s_keys[v - 1] > kv) { s_keys[v] = s_keys[v - 1]; --v; }
      s_keys[v] = kv;
    }
  }
  __syncthreads();

  // ---- Stage 6: mask = (argsort_output[j] < t), multiply, stage in LDS ---
  const unsigned t = (unsigned)total[row];
  float* s_outf = reinterpret_cast<float*>(s_hist);   // hist region is dead now
  for (int j = tid; j < KN; j += KT) {
    unsigned idx = (unsigned)(s_keys[j] & 0xFFFFFFFFull); // index of rank-j element
    s_outf[j] = s_vals[j] * ((idx < t) ? 1.0f : 0.0f);
  }
  __syncthreads();

  // ---- Stage 7: DMA the finished row back to HBM with the TDM ------------
#if __has_builtin(__builtin_amdgcn_tensor_store_from_lds)
  if (tid < 32) {
    u32x4 g0; i32x8 g1;
    tdm_row_desc(out + rbase, &s_outf[0], g0, g1);
    i32x4 z4 = {0, 0, 0, 0};
#if __clang_major__ >= 23
    i32x8 z8 = {0, 0, 0, 0, 0, 0, 0, 0};
    __builtin_amdgcn_tensor_store_from_lds(g0, g1, z4, z4, z8, 0);
#else
    __builtin_amdgcn_tensor_store_from_lds(g0, g1, z4, z4, 0);
#endif
    wait_tensorcnt0();   // keep LDS alive until the DMA has drained
  }
#else
  for (int j = tid; j < KN; j += KT) out[rbase + j] = s_outf[j];
#endif
}

extern "C" void kernel_launch(void* const* d_in, const int* in_sizes, int n_in,
                              void* d_out, int out_size, void* d_ws, size_t ws_size,
                              hipStream_t stream) {
  const float* proba = (const float*)d_in[0];
  const int*   total = (const int*)d_in[1];
  float*       out   = (float*)d_out;
  const int rows = in_sizes[1];   // B (in_total has B elements)
  hipLaunchKernelGGL(ProbaRanking_7834020348598_kernel,
                     dim3(rows), dim3(KT), 0, stream, proba, total, out);
}